// AbstractConv3D_57097295233505
// MI455X (gfx1250) — compile-verified
//
#include <hip/hip_runtime.h>

typedef __attribute__((ext_vector_type(2))) float v2f;
typedef __attribute__((ext_vector_type(8))) float v8f;

#define MS_NTOTAL 409600
#define MS_CIN 16
#define MS_COUT 16
#define MS_TILES_PER_BATCH (MS_NTOTAL / 16)   // 25600
#define MS_WAVES_PER_BLOCK 8

// Implicit-GEMM multi-resolution 3D conv using V_WMMA_F32_16X16X4_F32.
// One wave32 computes a [16 voxels x 16 Cout] tile, accumulating over
// 27 taps x 4 Cin-chunks = 108 WMMA ops. Halo handling is fully
// branchless: coordinates are clamped in-range (no EXEC divergence) and
// the A-fragment is multiplied by a 0/1 mask.
__global__ __launch_bounds__(256) void conv3d_ms_wmma_kernel(
    const float* __restrict__ x,     // [B, N_total, 16]
    const float* __restrict__ w,     // [16, 16, 3,3,3]
    const float* __restrict__ bias,  // [16]
    float* __restrict__ out)         // [B, N_total, 16]
{
    // B-fragment staging: wLds[((t*4 + c)*16 + n)*4 + k] = w[n][c*4+k][t]
    // so a lane's (t,c) fragment read is one contiguous 8-byte ds load.
    __shared__ float wLds[27 * 4 * 16 * 4];   // 6912 floats = 27 KB

    const int tid = threadIdx.x;
    for (int e = tid; e < 6912; e += 256) {
        const int k  = e & 3;
        const int n  = (e >> 2) & 15;
        const int tc = e >> 6;          // 0..107
        const int c  = tc & 3;
        const int t  = tc >> 2;
        wLds[e] = w[n * (16 * 27) + (c * 4 + k) * 27 + t];
    }
    __syncthreads();

    const int lane  = tid & 31;
    const int wave  = tid >> 5;
    const int n16   = lane & 15;   // Cout column / voxel row selector
    const int khalf = lane >> 4;   // 0: K=0,1   1: K=2,3  (A/B 32-bit layout)

    const int gt    = blockIdx.x * MS_WAVES_PER_BLOCK + wave;  // global tile
    const int b     = gt / MS_TILES_PER_BATCH;
    const int tloc  = gt - b * MS_TILES_PER_BATCH;
    const int n0    = tloc * 16 + n16;                          // voxel index

    // Level decode: offsets {0, 4096, 36864, 147456}, R in {16,32,48,64}.
    int R, s;
    if      (n0 < 4096)   { R = 16; s = 0;      }
    else if (n0 < 36864)  { R = 32; s = 4096;   }
    else if (n0 < 147456) { R = 48; s = 36864;  }
    else                  { R = 64; s = 147456; }
    const int l   = n0 - s;
    const int R2  = R * R;
    const int d0  = l / R2;
    const int rem = l - d0 * R2;
    const int d1  = rem / R;
    const int d2  = rem - d1 * R;
    const int Rm1 = R - 1;

    const float bv = bias[n16];
    v8f acc = {bv, bv, bv, bv, bv, bv, bv, bv};

    const float* xb = x + (long)b * (long)(MS_NTOTAL * MS_CIN);

#pragma unroll
    for (int t = 0; t < 27; ++t) {
        const int dz = t / 9 - 1;
        const int dy = (t / 3) % 3 - 1;
        const int dx = t % 3 - 1;
        const int z  = d0 + dz;
        const int y  = d1 + dy;
        const int xx = d2 + dx;
        const bool valid = ((unsigned)z  < (unsigned)R) &
                           ((unsigned)y  < (unsigned)R) &
                           ((unsigned)xx < (unsigned)R);
        // Clamp coordinates in-range: address math stays convergent (pure
        // v_min/v_max, no saveexec), mask-multiply produces the zero pad.
        const int zc = min(max(z, 0), Rm1);
        const int yc = min(max(y, 0), Rm1);
        const int xc = min(max(xx, 0), Rm1);
        const int nb = s + (zc * R + yc) * R + xc;
        const float msk = valid ? 1.0f : 0.0f;

        const float* px = xb + (long)nb * MS_CIN + khalf * 2;
        const float* pw = &wLds[t * 256 + n16 * 4 + khalf * 2];

#pragma unroll
        for (int c = 0; c < 4; ++c) {
            v2f a = *(const v2f*)(px + c * 4);   // A: 16x4 f32 fragment
            a = a * msk;
            v2f bf = *(const v2f*)(pw + c * 64); // B: 4x16 f32 fragment (LDS)
            acc = __builtin_amdgcn_wmma_f32_16x16x4_f32(
                false, a, false, bf, (short)0, acc, false, false);
        }
    }

    // D layout: lane L, vgpr j -> (m = j + 8*(L>>4), n = L&15)
    float* ob = out + ((long)b * MS_NTOTAL + (long)tloc * 16) * MS_COUT;
#pragma unroll
    for (int j = 0; j < 8; ++j) {
        const int m = j + khalf * 8;
        ob[m * MS_COUT + n16] = acc[j];
    }
}

extern "C" void kernel_launch(void* const* d_in, const int* in_sizes, int n_in,
                              void* d_out, int out_size, void* d_ws, size_t ws_size,
                              hipStream_t stream) {
    const float* x    = (const float*)d_in[0];  // [4, 409600, 16]
    const float* w    = (const float*)d_in[1];  // [16,16,3,3,3]
    const float* bias = (const float*)d_in[2];  // [16]
    float* out        = (float*)d_out;          // [4, 409600, 16]

    const int total_tiles = 4 * MS_TILES_PER_BATCH;           // 102400
    const int blocks      = total_tiles / MS_WAVES_PER_BLOCK; // 12800

    hipLaunchKernelGGL(conv3d_ms_wmma_kernel, dim3(blocks), dim3(256), 0, stream,
                       x, w, bias, out);
}